// LocalCovariance_1769526526730
// MI455X (gfx1250) — compile-verified
//
#include <hip/hip_runtime.h>

// LocalCovariance on MI455X (gfx1250):
//   d'_ij = -2 a_i . b_j + |b_j|^2   (row-constant |a_i|^2 dropped: top-K invariant)
//   computed as one V_WMMA_F32_16X16X4_F32 per 16x16 tile with
//   A_row = (x,y,z,1), B_col = (-2x,-2y,-2z,|b|^2).
// Top-16 selection streamed through LDS with per-lane register sorted lists,
// wave-wide argmin merge, then in-wave 3x3 covariance via shfl_xor reductions.

typedef __attribute__((ext_vector_type(2))) float v2f;
typedef __attribute__((ext_vector_type(8))) float v8f;

#define BATCH 16
#define NPTS  4096
#define KNN   16
#define COLS_PER_IT 256                 // 16 waves * 16 cols
#define NITER (NPTS / COLS_PER_IT)      // 16
#define LDS_STRIDE 257                  // 256 + 1 pad (bank-conflict avoidance)

__global__ __launch_bounds__(256)
void lc_augment_kernel(const float* __restrict__ x, float4* __restrict__ aug, int n) {
    int i = blockIdx.x * blockDim.x + threadIdx.x;
    if (i < n) {
        float a = x[3 * i + 0];
        float b = x[3 * i + 1];
        float c = x[3 * i + 2];
        aug[i] = make_float4(a, b, c, a * a + b * b + c * c);
    }
}

__global__ __launch_bounds__(512)
void lc_knn_cov_kernel(const float4* __restrict__ aug, float* __restrict__ out) {
    __shared__ float tile[16 * LDS_STRIDE];

    const int lane  = threadIdx.x & 31;
    const int w     = threadIdx.x >> 5;      // wave id 0..15, also the row this wave selects
    const int cloud = blockIdx.y;
    const int rowb  = blockIdx.x * 16;       // cloud-local row base of this block
    const int base  = cloud * NPTS;          // global point offset of this cloud
    const int n     = lane & 15;

    // ---- A operand (16x4 f32, ISA layout: lanes 0-15 hold K=0,1; lanes 16-31 hold K=2,3)
    float4 pa = aug[base + rowb + n];
    v2f avec;
    if (lane < 16) { avec.x = pa.x; avec.y = pa.y; }
    else           { avec.x = pa.z; avec.y = 1.0f; }

    // ---- per-lane sorted ascending top-16 (value,index) in registers
    const int row_local = rowb + w;
    float lv[KNN];
    int   li[KNN];
#pragma unroll
    for (int k = 0; k < KNN; ++k) { lv[k] = __builtin_inff(); li[k] = 0x7fffffff; }

    v8f czero = {};

    for (int it = 0; it < NITER; ++it) {
        // ---- B operand (4x16 f32): this wave's 16 columns
        int colb = it * COLS_PER_IT + w * 16;
        float4 qb = aug[base + colb + n];
        v2f bvec;
        if (lane < 16) { bvec.x = -2.0f * qb.x; bvec.y = -2.0f * qb.y; }
        else           { bvec.x = -2.0f * qb.z; bvec.y = qb.w; }

        // ---- one WMMA = full 16x16 fp32 distance tile
        v8f d = __builtin_amdgcn_wmma_f32_16x16x4_f32(
            /*neg_a=*/false, avec, /*neg_b=*/false, bvec,
            /*c_mod=*/(short)0, czero, /*reuse_a=*/false, /*reuse_b=*/false);

        // ---- scatter C tile to LDS (VGPR r: lanes 0-15 -> M=r, lanes 16-31 -> M=r+8)
        int rb  = (lane < 16) ? 0 : 8;
        int col = w * 16 + n;
#pragma unroll
        for (int r = 0; r < 8; ++r)
            tile[(r + rb) * LDS_STRIDE + col] = d[r];

        __syncthreads();

        // ---- selection: wave w scans row w's 256 fresh columns (8 per lane)
        const float* rowp = &tile[w * LDS_STRIDE];
#pragma unroll
        for (int k = 0; k < 8; ++k) {
            int   c  = lane + k * 32;
            float dv = rowp[c];
            int   j  = it * COLS_PER_IT + c;          // cloud-local neighbor index
            if (j == row_local) dv = __builtin_inff(); // exclude self-edge
            if (dv < lv[KNN - 1]) {
                lv[KNN - 1] = dv; li[KNN - 1] = j;
#pragma unroll
                for (int s = KNN - 1; s > 0; --s) {    // single bubble pass (list was sorted)
                    if (lv[s] < lv[s - 1]) {
                        float tv = lv[s]; lv[s] = lv[s - 1]; lv[s - 1] = tv;
                        int   ti = li[s]; li[s] = li[s - 1]; li[s - 1] = ti;
                    }
                }
            }
        }
        __syncthreads();
    }

    // ---- merge 32 per-lane sorted lists -> global top-16 (lane r keeps r-th neighbor)
    int my_nbr = 0;
#pragma unroll
    for (int r = 0; r < KNN; ++r) {
        float v = lv[0];
        int   i = li[0];
#pragma unroll
        for (int m = 1; m <= 16; m <<= 1) {
            float ov = __shfl_xor(v, m, 32);
            int   oi = __shfl_xor(i, m, 32);
            if (ov < v || (ov == v && oi < i)) { v = ov; i = oi; }
        }
        if (lane == r) my_nbr = i;
        if (li[0] == i) {                    // unique winner pops its head
#pragma unroll
            for (int s = 0; s < KNN - 1; ++s) { lv[s] = lv[s + 1]; li[s] = li[s + 1]; }
            lv[KNN - 1] = __builtin_inff();
            li[KNN - 1] = 0x7fffffff;
        }
    }

    // ---- gather 16 neighbors (lanes 0-15) and reduce mean + covariance in-wave
    float px = 0.0f, py = 0.0f, pz = 0.0f;
    if (lane < 16) {
        float4 q = aug[base + my_nbr];
        px = q.x; py = q.y; pz = q.z;
    }
    float sx = px, sy = py, sz = pz;
#pragma unroll
    for (int m = 1; m <= 8; m <<= 1) {
        sx += __shfl_xor(sx, m, 32);
        sy += __shfl_xor(sy, m, 32);
        sz += __shfl_xor(sz, m, 32);
    }
    const float inv = 1.0f / 16.0f;
    float mx = sx * inv, my = sy * inv, mz = sz * inv;
    float cx = px - mx, cy = py - my, cz = pz - mz;
    float xx = cx * cx, xy = cx * cy, xz = cx * cz;
    float yy = cy * cy, yz = cy * cz, zz = cz * cz;
#pragma unroll
    for (int m = 1; m <= 8; m <<= 1) {
        xx += __shfl_xor(xx, m, 32); xy += __shfl_xor(xy, m, 32); xz += __shfl_xor(xz, m, 32);
        yy += __shfl_xor(yy, m, 32); yz += __shfl_xor(yz, m, 32); zz += __shfl_xor(zz, m, 32);
    }
    if (lane == 0) {
        float4 q = aug[base + row_local];
        float* o = out + (size_t)(base + row_local) * 12;
        o[0]  = q.x;      o[1]  = q.y;      o[2]  = q.z;
        o[3]  = xx * inv; o[4]  = xy * inv; o[5]  = xz * inv;
        o[6]  = xy * inv; o[7]  = yy * inv; o[8]  = yz * inv;
        o[9]  = xz * inv; o[10] = yz * inv; o[11] = zz * inv;
    }
}

extern "C" void kernel_launch(void* const* d_in, const int* in_sizes, int n_in,
                              void* d_out, int out_size, void* d_ws, size_t ws_size,
                              hipStream_t stream) {
    const float* x   = (const float*)d_in[0];   // (B*P, 3) float32
    float*       out = (float*)d_out;           // (B*P, 12) float32
    float4*      aug = (float4*)d_ws;           // B*P float4 = 1 MB scratch

    const int npts = BATCH * NPTS;              // 65536
    lc_augment_kernel<<<dim3((npts + 255) / 256), dim3(256), 0, stream>>>(x, aug, npts);

    dim3 grid(NPTS / 16, BATCH);                // 256 x 16 workgroups
    lc_knn_cov_kernel<<<grid, dim3(512), 0, stream>>>(aug, out);
}